// BPHDNNP_61435212202592
// MI455X (gfx1250) — compile-verified
//
#include <hip/hip_runtime.h>

// ---------------------------------------------------------------------------
// BP-HDNNP fused kernel for MI455X (gfx1250, wave32, WMMA)
//   X:[B,N,128] f32, Z:[B,N] i32 in [0,4)
//   per-element MLP 128 ->(tanh) 64 ->(tanh) 64 -> 1, sum over N per molecule
// One workgroup = 256 atoms = 4 molecules. All 4 element-MLPs' weights staged
// in LDS (~194 KB of the 320 KB WGP pool) in a K-pair-interleaved layout so
// every WMMA B operand is a single contiguous ds_load_b64. Atoms are binned
// by element id; 16-atom tiles run through V_WMMA_F32_16X16X4_F32 in f32.
// ---------------------------------------------------------------------------

typedef __attribute__((ext_vector_type(2))) float v2f;
typedef __attribute__((ext_vector_type(8))) float v8f;

#if defined(__has_builtin)
#  if __has_builtin(__builtin_amdgcn_tanhf)
#    define TANHF(x) __builtin_amdgcn_tanhf(x)   // hardware V_TANH_F32
#  else
#    define TANHF(x) tanhf(x)
#  endif
#else
#  define TANHF(x) tanhf(x)
#endif

#define NELEM   4
#define DESC    128
#define HID     64
#define CHUNK   256          // atoms per workgroup (= 4 molecules of 64 atoms)
#define NWAVES  8
#define HSTRIDE 66           // 16x64 staging row stride: even (b64-aligned), banks spread

// LDS float-region sizes
#define W1_F  (NELEM*DESC*HID)        // 32768 floats = 128 KB
#define W2_F  (NELEM*HID*HID)         // 16384 floats =  64 KB
#define W3_F  (NELEM*HID)             //   256
#define B1_F  (NELEM*HID)             //   256
#define B2_F  (NELEM*HID)             //   256
#define B3_F  (NELEM)                 //     4
#define HST_F (NWAVES*16*HSTRIDE)     //  8448
#define MOLE_F 4
#define FLOATS_TOTAL (W1_F+W2_F+W3_F+B1_F+B2_F+B3_F+HST_F+MOLE_F)
#define INTS_TOTAL   (NELEM + NELEM*CHUNK)     // bin counts + bin index lists
#define SMEM_BYTES   ((size_t)(FLOATS_TOTAL + INTS_TOTAL) * 4)

__global__ __launch_bounds__(256)
void hdnnp_wmma_kernel(const float* __restrict__ X,
                       const int*   __restrict__ Z,
                       const float* __restrict__ W1,
                       const float* __restrict__ b1,
                       const float* __restrict__ W2,
                       const float* __restrict__ b2,
                       const float* __restrict__ W3,
                       const float* __restrict__ b3,
                       float*       __restrict__ out)
{
    extern __shared__ float smem[];
    // W1/W2 stored K-pair interleaved: [e][kk][half][n][2] where the last
    // index packs K = 4*kk + 2*half + {0,1}.  B operand = contiguous v2f.
    float* w1s    = smem;                         // [E][32][2][64][2]
    float* w2s    = w1s + W1_F;                   // [E][16][2][64][2]
    float* w3s    = w2s + W2_F;                   // [E][64]
    float* b1s    = w3s + W3_F;                   // [E][64]
    float* b2s    = b1s + B1_F;                   // [E][64]
    float* b3s    = b2s + B2_F;                   // [E]
    float* hstage = b3s + B3_F;                   // per-wave 16 x HSTRIDE staging
    float* molE   = hstage + HST_F;               // [4] molecule accumulators
    int*   cnt    = (int*)(molE + MOLE_F);        // [E] bin counts
    int*   binIdx = cnt + NELEM;                  // [E][256] slot lists

    const int tid  = threadIdx.x;
    const int lane = tid & 31;
    const int wave = tid >> 5;
    const int m    = lane & 15;                   // row / column within tile
    const int hf   = lane >> 4;                   // lane half (K split)
    const int base = blockIdx.x * CHUNK;          // first atom of this chunk

    if (tid < NELEM) cnt[tid] = 0;
    if (tid < 4)     molE[tid] = 0.0f;
    __syncthreads();

    // ---- bin atoms by element id (LDS atomic append) ----
    {
        int z   = Z[base + tid];
        int pos = atomicAdd(&cnt[z], 1);
        binIdx[z * CHUNK + pos] = tid;
    }

    // ---- cooperative weight staging with K-pair interleave permutation ----
    {
        const float4* s1 = (const float4*)W1;     // W1[e][d][n], n fastest
        for (int i = tid; i < W1_F / 4; i += 256) {
            float4 v = s1[i];
            int g  = i * 4;
            int n  = g & (HID - 1);
            int d  = (g >> 6) & (DESC - 1);
            int e  = g >> 13;                     // g / (DESC*HID)
            int kk = d >> 2, hh = (d >> 1) & 1, lo = d & 1;
            float* dst = w1s + ((e * (DESC/4) + kk) * 2 + hh) * (2*HID) + 2*n + lo;
            dst[0] = v.x; dst[2] = v.y; dst[4] = v.z; dst[6] = v.w;
        }
        const float4* s2 = (const float4*)W2;     // W2[e][h][n], n fastest
        for (int i = tid; i < W2_F / 4; i += 256) {
            float4 v = s2[i];
            int g  = i * 4;
            int n  = g & (HID - 1);
            int h  = (g >> 6) & (HID - 1);
            int e  = g >> 12;                     // g / (HID*HID)
            int kk = h >> 2, hh = (h >> 1) & 1, lo = h & 1;
            float* dst = w2s + ((e * (HID/4) + kk) * 2 + hh) * (2*HID) + 2*n + lo;
            dst[0] = v.x; dst[2] = v.y; dst[4] = v.z; dst[6] = v.w;
        }
        if (tid < W3_F) w3s[tid] = W3[tid];
        if (tid < B1_F) b1s[tid] = b1[tid];
        if (tid < B2_F) b2s[tid] = b2[tid];
        if (tid < B3_F) b3s[tid] = b3[tid];
    }
    __syncthreads();

    // ---- tile schedule: padded-to-16 tiles over the 4 bins, round-robin ----
    const int c0 = cnt[0], c1 = cnt[1], c2 = cnt[2], c3 = cnt[3];
    const int t0 = (c0 + 15) >> 4, t1 = (c1 + 15) >> 4;
    const int t2 = (c2 + 15) >> 4, t3 = (c3 + 15) >> 4;
    const int T  = t0 + t1 + t2 + t3;

    float* hst = hstage + wave * (16 * HSTRIDE);

    for (int t = wave; t < T; t += NWAVES) {
        // map flat tile id -> (element e, tile-in-bin lt), wave-uniform
        int e, lt, r = t;
        if (r < t0)            { e = 0; lt = r; }
        else { r -= t0;
        if (r < t1)            { e = 1; lt = r; }
        else { r -= t1;
        if (r < t2)            { e = 2; lt = r; }
        else                   { e = 3; lt = r - t2; } } }
        const int ccnt = cnt[e];

        const int  pos   = lt * 16 + m;
        const bool valid = pos < ccnt;
        const int  slot  = valid ? binIdx[e * CHUNK + pos] : 0;   // pad -> atom 0
        // A-operand source row: lane L and L+16 read the same atom row,
        // offset by 2*hf to cover K = {0,1} / {2,3} of each 4-deep k-step.
        const float* xrow = X + (size_t)(base + slot) * DESC + 2 * hf;

        // -------- layer 1: h1 = tanh(x @ W1[e] + b1[e]),  K = 128 --------
        v8f a0 = {}, a1 = {}, a2 = {}, a3 = {};
        const float* w1e = w1s + e * (DESC * HID);
        #pragma unroll 4
        for (int kk = 0; kk < DESC / 4; ++kk) {
            v2f a = *(const v2f*)(xrow + 4 * kk);                 // global b64 gather
            const float* wb = w1e + (kk * 2 + hf) * (2*HID) + 2*m;
            v2f bt0 = *(const v2f*)(wb);                          // ds_load_b64
            v2f bt1 = *(const v2f*)(wb + 32);
            v2f bt2 = *(const v2f*)(wb + 64);
            v2f bt3 = *(const v2f*)(wb + 96);
            a0 = __builtin_amdgcn_wmma_f32_16x16x4_f32(false, a, false, bt0, (short)0, a0, false, false);
            a1 = __builtin_amdgcn_wmma_f32_16x16x4_f32(false, a, false, bt1, (short)0, a1, false, false);
            a2 = __builtin_amdgcn_wmma_f32_16x16x4_f32(false, a, false, bt2, (short)0, a2, false, false);
            a3 = __builtin_amdgcn_wmma_f32_16x16x4_f32(false, a, false, bt3, (short)0, a3, false, false);
        }
        {   // bias + tanh, transpose through LDS staging tile
            const float* b1e = b1s + e * HID;
            const float z0 = b1e[m], z1 = b1e[16 + m], z2 = b1e[32 + m], z3 = b1e[48 + m];
            #pragma unroll
            for (int g = 0; g < 8; ++g) {
                float* hr = hst + (g + 8 * hf) * HSTRIDE + m;     // C/D: M = g + 8*half
                hr[0]  = TANHF(a0[g] + z0);
                hr[16] = TANHF(a1[g] + z1);
                hr[32] = TANHF(a2[g] + z2);
                hr[48] = TANHF(a3[g] + z3);
            }
        }

        // -------- layer 2: h2 = tanh(h1 @ W2[e] + b2[e]),  K = 64 --------
        v8f d0 = {}, d1 = {}, d2 = {}, d3 = {};
        const float* w2e  = w2s + e * (HID * HID);
        const float* arow = hst + m * HSTRIDE + 2 * hf;           // A from staging
        #pragma unroll 4
        for (int kk = 0; kk < HID / 4; ++kk) {
            v2f a = *(const v2f*)(arow + 4 * kk);                 // ds_load_b64 (aligned)
            const float* wb = w2e + (kk * 2 + hf) * (2*HID) + 2*m;
            v2f bt0 = *(const v2f*)(wb);
            v2f bt1 = *(const v2f*)(wb + 32);
            v2f bt2 = *(const v2f*)(wb + 64);
            v2f bt3 = *(const v2f*)(wb + 96);
            d0 = __builtin_amdgcn_wmma_f32_16x16x4_f32(false, a, false, bt0, (short)0, d0, false, false);
            d1 = __builtin_amdgcn_wmma_f32_16x16x4_f32(false, a, false, bt1, (short)0, d1, false, false);
            d2 = __builtin_amdgcn_wmma_f32_16x16x4_f32(false, a, false, bt2, (short)0, d2, false, false);
            d3 = __builtin_amdgcn_wmma_f32_16x16x4_f32(false, a, false, bt3, (short)0, d3, false, false);
        }
        {   // bias + tanh, write h2 back into staging (layer-2 reads are done)
            const float* b2e = b2s + e * HID;
            const float z0 = b2e[m], z1 = b2e[16 + m], z2 = b2e[32 + m], z3 = b2e[48 + m];
            #pragma unroll
            for (int g = 0; g < 8; ++g) {
                float* hr = hst + (g + 8 * hf) * HSTRIDE + m;
                hr[0]  = TANHF(d0[g] + z0);
                hr[16] = TANHF(d1[g] + z1);
                hr[32] = TANHF(d2[g] + z2);
                hr[48] = TANHF(d3[g] + z3);
            }
        }

        // -------- layer 3: e = h2 . W3[e] + b3[e], pool into molecule --------
        {
            float en = 0.0f;
            const float* w3e = w3s + e * HID;
            const float* hr  = hst + m * HSTRIDE;                 // row of atom m
            #pragma unroll 8
            for (int h = 0; h < HID; ++h) en = fmaf(hr[h], w3e[h], en);
            en += b3s[e];
            if (hf == 0 && valid)
                atomicAdd(&molE[slot >> 6], en);                  // ds_add_f32
        }
    }

    __syncthreads();
    if (tid < 4) out[blockIdx.x * 4 + tid] = molE[tid];
}

// ---------------------------------------------------------------------------
extern "C" void kernel_launch(void* const* d_in, const int* in_sizes, int n_in,
                              void* d_out, int out_size, void* d_ws, size_t ws_size,
                              hipStream_t stream) {
    (void)n_in; (void)out_size; (void)d_ws; (void)ws_size;
    const float* X  = (const float*)d_in[0];
    const int*   Z  = (const int*)  d_in[1];
    const float* W1 = (const float*)d_in[2];
    const float* b1 = (const float*)d_in[3];
    const float* W2 = (const float*)d_in[4];
    const float* b2 = (const float*)d_in[5];
    const float* W3 = (const float*)d_in[6];
    const float* b3 = (const float*)d_in[7];
    float* out = (float*)d_out;

    const int M      = in_sizes[1];        // B*N atoms (Z element count)
    const int blocks = M / CHUNK;          // 4 molecules (256 atoms) per block

    hipLaunchKernelGGL(hdnnp_wmma_kernel, dim3(blocks), dim3(256), SMEM_BYTES,
                       stream, X, Z, W1, b1, W2, b2, W3, b3, out);
}